// BatchMultiHeadGraphAttention_14989435863223
// MI455X (gfx1250) — compile-verified
//
#include <hip/hip_runtime.h>
#include <hip/hip_bf16.h>

typedef __attribute__((ext_vector_type(2))) float v2f;
typedef __attribute__((ext_vector_type(8))) float v8f;

#define BS 32
#define NN 512
#define NH 8
#define FF 64
#define NEG_SLOPE 0.2f
#define KPAD 516   // P row stride: 516%64==4 -> conflict-free A-frag b64 reads; rows 16B-aligned
#define SPAD 68    // stage row stride: 68*4 = 272 B, multiple of 16 -> B128-aligned rows

// Branchless tanh via v_exp + v_rcp: 1 - 2*rcp(e^{2x}+1). Saturates to +-1.
__device__ __forceinline__ float fast_tanh(float x) {
  return 1.0f - 2.0f * __builtin_amdgcn_rcpf(__expf(2.0f * x) + 1.0f);
}

// CDNA5 async copy: global -> LDS, 16 bytes per active lane, tracked by ASYNCcnt.
__device__ __forceinline__ void async_copy_b128(unsigned lds_off, unsigned long long gaddr) {
  asm volatile("global_load_async_to_lds_b128 %0, %1, off"
               :: "v"(lds_off), "v"(gaddr) : "memory");
}
__device__ __forceinline__ void wait_async0() {
  asm volatile("s_wait_asynccnt 0x0" ::: "memory");
}

// ---------------------------------------------------------------------------
// Kernel 1: h_prime[b,h] = h[b] (512x64) @ w[h] (64x64), plus fused
//           attn_src[b,h,n] = sum_o tanh(h_prime) * a_src[h,o]  (same for dst)
// Grid: BS*NH blocks, 256 threads (8 waves). f32 WMMA 16x16x4, K=64.
// ---------------------------------------------------------------------------
__global__ __launch_bounds__(256) void gat_hprime_kernel(
    const float* __restrict__ h, const float* __restrict__ wmat,
    const float* __restrict__ a_src, const float* __restrict__ a_dst,
    float* __restrict__ hp, float* __restrict__ s_src, float* __restrict__ s_dst)
{
  const int bh   = blockIdx.x;
  const int b    = bh / NH;
  const int hd   = bh % NH;
  const int wave = threadIdx.x >> 5;
  const int lane = threadIdx.x & 31;
  const int l15  = lane & 15;
  const int hi   = lane >> 4;   // 0: lanes 0-15 (M=0..7 / K=0,1), 1: lanes 16-31 (M=8..15 / K=2,3)

  const float* __restrict__ hB  = h    + (size_t)b  * NN * FF;
  const float* __restrict__ wH  = wmat + (size_t)hd * FF * FF;
  const float* __restrict__ asH = a_src + (size_t)hd * FF;
  const float* __restrict__ adH = a_dst + (size_t)hd * FF;
  float* __restrict__ hpBH = hp + (size_t)bh * NN * FF;

  for (int rs = 0; rs < 4; ++rs) {
    const int r0 = (rs * 8 + wave) * 16;
    v8f acc0 = {}, acc1 = {}, acc2 = {}, acc3 = {};
    const int arow = r0 + l15;

#pragma unroll 4
    for (int kk = 0; kk < 16; ++kk) {
      const int kb = kk * 4 + hi * 2;
      v2f a;                                   // A frag (16x4)
      a.x = hB[arow * FF + kb];
      a.y = hB[arow * FF + kb + 1];
      v2f bv0, bv1, bv2, bv3;                  // B frags (4x16), 4 col tiles
      bv0.x = wH[kb * FF + (l15 +  0)]; bv0.y = wH[(kb + 1) * FF + (l15 +  0)];
      bv1.x = wH[kb * FF + (l15 + 16)]; bv1.y = wH[(kb + 1) * FF + (l15 + 16)];
      bv2.x = wH[kb * FF + (l15 + 32)]; bv2.y = wH[(kb + 1) * FF + (l15 + 32)];
      bv3.x = wH[kb * FF + (l15 + 48)]; bv3.y = wH[(kb + 1) * FF + (l15 + 48)];

      acc0 = __builtin_amdgcn_wmma_f32_16x16x4_f32(false, a, false, bv0, (short)0, acc0, false, false);
      acc1 = __builtin_amdgcn_wmma_f32_16x16x4_f32(false, a, false, bv1, (short)0, acc1, false, false);
      acc2 = __builtin_amdgcn_wmma_f32_16x16x4_f32(false, a, false, bv2, (short)0, acc2, false, false);
      acc3 = __builtin_amdgcn_wmma_f32_16x16x4_f32(false, a, false, bv3, (short)0, acc3, false, false);
    }

    // Epilogue (fully unrolled): store h_prime, fused tanh dot + half-wave reduce.
    const int m0 = r0 + hi * 8;
    const float as0 = asH[l15 +  0], as1 = asH[l15 + 16], as2 = asH[l15 + 32], as3 = asH[l15 + 48];
    const float ad0 = adH[l15 +  0], ad1 = adH[l15 + 16], ad2 = adH[l15 + 32], ad3 = adH[l15 + 48];

#pragma unroll
    for (int v = 0; v < 8; ++v) {
      const int m = m0 + v;
      const float x0 = acc0[v], x1 = acc1[v], x2 = acc2[v], x3 = acc3[v];
      hpBH[m * FF + l15 +  0] = x0;
      hpBH[m * FF + l15 + 16] = x1;
      hpBH[m * FF + l15 + 32] = x2;
      hpBH[m * FF + l15 + 48] = x3;
      const float t0 = fast_tanh(x0), t1 = fast_tanh(x1);
      const float t2 = fast_tanh(x2), t3 = fast_tanh(x3);
      float ps = t0 * as0 + t1 * as1 + t2 * as2 + t3 * as3;
      float pd = t0 * ad0 + t1 * ad1 + t2 * ad2 + t3 * ad3;
#pragma unroll
      for (int off = 1; off < 16; off <<= 1) {   // offsets < 16 stay within each half-wave
        ps += __shfl_xor(ps, off, 32);
        pd += __shfl_xor(pd, off, 32);
      }
      if (l15 == 0) {
        s_src[(size_t)bh * NN + m] = ps;
        s_dst[(size_t)bh * NN + m] = pd;
      }
    }
  }
}

// ---------------------------------------------------------------------------
// Kernel 2: per (b,h, 16-row strip): masked leaky-relu scores -> softmax
// numerators in LDS -> P(16x512) @ h_prime(512x64) via WMMA, normalize + bias.
// h_prime is streamed into LDS in 64-row chunks with global_load_async_to_lds.
// Grid: BS*NH*(NN/16) blocks, 128 threads (4 waves).
// ---------------------------------------------------------------------------
__global__ __launch_bounds__(128) void gat_attn_kernel(
    const unsigned char* __restrict__ adj, const float* __restrict__ hp,
    const float* __restrict__ s_src, const float* __restrict__ s_dst,
    const float* __restrict__ bias, float* __restrict__ out)
{
  __shared__ float P[16 * KPAD];        // softmax numerators, 16 rows x 512
  __shared__ float stage[64 * SPAD];    // async-staged h_prime chunk, 64 rows x 64
  __shared__ float rinv[16];            // reciprocal row sums

  const int blk  = blockIdx.x;
  const int it   = blk & 31;            // row tile index within (b,h)
  const int bh   = blk >> 5;            // b*NH + h
  const int b    = bh / NH;
  const int wave = threadIdx.x >> 5;
  const int lane = threadIdx.x & 31;
  const int tid  = threadIdx.x;
  const int i0   = it * 16;

  const float* __restrict__ srcV = s_src + (size_t)bh * NN;
  const float* __restrict__ dstV = s_dst + (size_t)bh * NN;
  const float* __restrict__ hpBH = hp + (size_t)bh * NN * FF;

  // Phase A: each wave computes softmax numerators for 4 rows.
  // Each lane owns 16 CONTIGUOUS columns -> adj is one b128 load, dst is 4x b128.
  const int jb = lane * 16;
#pragma unroll
  for (int rr = 0; rr < 4; ++rr) {
    const int li = wave * 4 + rr;
    const int i  = i0 + li;
    const float si = srcV[i];
    const unsigned char* __restrict__ adjRow = adj + ((size_t)b * NN + i) * NN;

    const uint4  aw = *(const uint4*)(adjRow + jb);          // 16 adjacency bytes
    const float4 d0 = *(const float4*)(dstV + jb +  0);
    const float4 d1 = *(const float4*)(dstV + jb +  4);
    const float4 d2 = *(const float4*)(dstV + jb +  8);
    const float4 d3 = *(const float4*)(dstV + jb + 12);
    float dv[16] = { d0.x, d0.y, d0.z, d0.w,  d1.x, d1.y, d1.z, d1.w,
                     d2.x, d2.y, d2.z, d2.w,  d3.x, d3.y, d3.z, d3.w };
    unsigned mw[4] = { aw.x, aw.y, aw.z, aw.w };

    float ev[16];
    float mx = -3.0e38f;
#pragma unroll
    for (int s = 0; s < 16; ++s) {
      float e = si + dv[s];
      e = (e >= 0.f) ? e : NEG_SLOPE * e;
      const bool on = ((mw[s >> 2] >> ((s & 3) * 8)) & 0xffu) != 0u;
      ev[s] = on ? e : -3.0e38f;
      mx = fmaxf(mx, ev[s]);
    }
#pragma unroll
    for (int off = 1; off < 32; off <<= 1) mx = fmaxf(mx, __shfl_xor(mx, off, 32));

    float ssum = 0.f;
#pragma unroll
    for (int s = 0; s < 16; ++s) {
      const float p = (ev[s] > -1.0e38f) ? __expf(ev[s] - mx) : 0.f;
      P[li * KPAD + jb + s] = p;                // 16B-aligned runs -> ds_store_b128
      ssum += p;
    }
#pragma unroll
    for (int off = 1; off < 32; off <<= 1) ssum += __shfl_xor(ssum, off, 32);
    if (lane == 0) rinv[li] = __builtin_amdgcn_rcpf(ssum);
  }
  __syncthreads();

  // Phase B: K=512 WMMA loop in 8 chunks of 64 rows; each chunk async-staged
  // into LDS as 1024 coalesced B128 transfers (8 per thread).
  const int l15 = lane & 15;
  const int hi  = lane >> 4;
  const int c0  = wave * 16;
  v8f acc = {};

  for (int kc = 0; kc < 8; ++kc) {
    const float* __restrict__ gsrc = hpBH + (size_t)kc * 64 * FF;
#pragma unroll
    for (int q = 0; q < 8; ++q) {
      const int idx = q * 128 + tid;           // 0..1023
      const int r   = idx >> 4;                // row 0..63
      const int s4  = (idx & 15) * 4;          // 16B segment within row
      const unsigned lds_off = (unsigned)(uintptr_t)(stage + r * SPAD + s4);
      const unsigned long long ga = (unsigned long long)(uintptr_t)(gsrc + r * FF + s4);
      async_copy_b128(lds_off, ga);
    }
    wait_async0();
    __syncthreads();

#pragma unroll 4
    for (int kkl = 0; kkl < 16; ++kkl) {
      const int kb = kkl * 4 + hi * 2;         // local k within chunk
      const int kg = kc * 64 + kb;             // global k for P
      v2f a, bm;
      a.x  = P[l15 * KPAD + kg];
      a.y  = P[l15 * KPAD + kg + 1];
      bm.x = stage[kb       * SPAD + c0 + l15];
      bm.y = stage[(kb + 1) * SPAD + c0 + l15];
      acc = __builtin_amdgcn_wmma_f32_16x16x4_f32(false, a, false, bm, (short)0, acc, false, false);
    }
    __syncthreads();   // protect stage before next chunk overwrites it
  }

  const int m0 = hi * 8;
#pragma unroll
  for (int v = 0; v < 8; ++v) {
    const int lm = m0 + v;
    const float o = acc[v] * rinv[lm] + bias[c0 + l15];
    out[((size_t)bh * NN + (i0 + lm)) * FF + c0 + l15] = o;
  }
}

// ---------------------------------------------------------------------------
extern "C" void kernel_launch(void* const* d_in, const int* in_sizes, int n_in,
                              void* d_out, int out_size, void* d_ws, size_t ws_size,
                              hipStream_t stream) {
  const float*         h     = (const float*)d_in[0];
  const unsigned char* adj   = (const unsigned char*)d_in[1];  // jax bool -> 1 byte
  const float*         wmat  = (const float*)d_in[2];
  const float*         a_src = (const float*)d_in[3];
  const float*         a_dst = (const float*)d_in[4];
  const float*         bias  = (const float*)d_in[5];
  float* out = (float*)d_out;

  // workspace: h_prime (32 MB) | attn_src (0.5 MB) | attn_dst (0.5 MB)
  float* hpw  = (float*)d_ws;
  float* ssrc = hpw + (size_t)BS * NH * NN * FF;
  float* sdst = ssrc + (size_t)BS * NH * NN;

  gat_hprime_kernel<<<BS * NH, 256, 0, stream>>>(h, wmat, a_src, a_dst, hpw, ssrc, sdst);
  gat_attn_kernel<<<BS * NH * (NN / 16), 128, 0, stream>>>(adj, hpw, ssrc, sdst, bias, out);
}